// MyModel_61933428412730
// MI455X (gfx1250) — compile-verified
//
#include <hip/hip_runtime.h>
#include <hip/hip_bf16.h>

// Pairwise L1 distance: x[8192,128] f32 -> out[8192,8192] f32.
// Compute-bound (17.2 G VALU ops vs ~11us of HBM store traffic), so the
// design goal is max fp32 VALU throughput + CDNA5 async-to-LDS data movement.

#define NROWS 8192
#define DIM   128
#define BT    128          // output tile edge per block
#define DC    16           // d-chunk staged in LDS per step
#define LS    (DC + 4)     // padded LDS row stride (floats): 20 -> conflict-free B reads
#define NCHUNK (DIM / DC)  // 8
#define TILE_FLOATS (BT * LS)          // one tile-chunk in LDS
#define BUF_FLOATS  (2 * TILE_FLOATS)  // A + B tiles per buffer

#if defined(__gfx1250__) && \
    __has_builtin(__builtin_amdgcn_global_load_async_to_lds_b128) && \
    __has_builtin(__builtin_amdgcn_s_wait_asynccnt)
#define USE_ASYNC_LDS 1
#else
#define USE_ASYNC_LDS 0
#endif

// Builtin signature (per hipcc diagnostic): pointee is a GCC-style vector of
// 4 ints; global side in AS1, LDS side in AS3.
typedef int v4i __attribute__((vector_size(4 * sizeof(int))));
typedef __attribute__((address_space(1))) v4i glb_v4i;
typedef __attribute__((address_space(3))) v4i lds_v4i;

__device__ __forceinline__ lds_v4i* to_lds(const void* p) {
  // generic LDS address: low 32 bits are the LDS byte offset (ISA 10.2)
  return (lds_v4i*)(unsigned int)(unsigned long long)p;
}
__device__ __forceinline__ glb_v4i* to_glb(const void* p) {
  return (glb_v4i*)(unsigned long long)p;
}

// Issue the async copy of one d-chunk (A tile + B tile) into `buf`.
// 2 tiles * 128 rows * 16 floats = 1024 float4 slots -> 4 per thread.
__device__ __forceinline__ void issue_chunk(const float* __restrict__ x,
                                            int tid, int rowBaseA, int rowBaseB,
                                            int dBase, float* buf) {
#pragma unroll
  for (int k = 0; k < 4; ++k) {
    int e    = tid + 256 * k;        // 0..1023
    int tile = e >> 9;               // 0 = A, 1 = B (512 float4 per tile)
    int rr   = (e >> 2) & (BT - 1);  // row within tile
    int seg  = e & 3;                // 16B segment within the d-chunk
    int grow = (tile ? rowBaseB : rowBaseA) + rr;
    const float* gp = x + (size_t)grow * DIM + dBase + seg * 4;
    float*       lp = buf + tile * TILE_FLOATS + rr * LS + seg * 4;
#if USE_ASYNC_LDS
    __builtin_amdgcn_global_load_async_to_lds_b128(to_glb(gp), to_lds(lp), 0, 0);
#else
    *(float4*)lp = *(const float4*)gp;
#endif
  }
}

__device__ __forceinline__ void wait_chunk() {
#if USE_ASYNC_LDS
  __builtin_amdgcn_s_wait_asynccnt(0);
#endif
  __syncthreads();
}

__global__ __launch_bounds__(256)
void pairwise_l1_kernel(const float* __restrict__ x, float* __restrict__ out) {
  __shared__ float lds[2 * BUF_FLOATS];  // 40 KB: double-buffered A+B tiles

  const int tid = threadIdx.x;
  const int tx  = tid & 15;  // -> output col j = colBase + tx + 16c
  const int ty  = tid >> 4;  // -> output row i = rowBase + ty + 16r
  const int rowBase = blockIdx.y * BT;
  const int colBase = blockIdx.x * BT;

  float acc[8][8] = {};

  // stage chunk 0
  issue_chunk(x, tid, rowBase, colBase, 0, &lds[0]);

#pragma unroll
  for (int ch = 0; ch < NCHUNK; ++ch) {
    wait_chunk();  // own async loads done + all waves arrived => chunk ch is in LDS

    if (ch + 1 < NCHUNK)  // prefetch next chunk into the other buffer
      issue_chunk(x, tid, rowBase, colBase, (ch + 1) * DC,
                  &lds[((ch + 1) & 1) * BUF_FLOATS]);

    const float* A = &lds[(ch & 1) * BUF_FLOATS];
    const float* B = A + TILE_FLOATS;

#pragma unroll
    for (int d = 0; d < DC; d += 4) {
      float4 a[8], b[8];
#pragma unroll
      for (int r = 0; r < 8; ++r)
        a[r] = *(const float4*)&A[(ty + 16 * r) * LS + d];
#pragma unroll
      for (int c = 0; c < 8; ++c)
        b[c] = *(const float4*)&B[(tx + 16 * c) * LS + d];
#pragma unroll
      for (int r = 0; r < 8; ++r)
#pragma unroll
        for (int c = 0; c < 8; ++c) {
          acc[r][c] += __builtin_fabsf(a[r].x - b[c].x)
                     + __builtin_fabsf(a[r].y - b[c].y)
                     + __builtin_fabsf(a[r].z - b[c].z)
                     + __builtin_fabsf(a[r].w - b[c].w);
        }
    }
    __syncthreads();  // everyone done reading buf[ch&1] before it is refilled
  }

  // Streaming 256 MB result: non-temporal stores keep the input hot in L2.
#pragma unroll
  for (int r = 0; r < 8; ++r) {
    size_t rowOff = (size_t)(rowBase + ty + 16 * r) * NROWS + colBase;
#pragma unroll
    for (int c = 0; c < 8; ++c)
      __builtin_nontemporal_store(acc[r][c], out + rowOff + tx + 16 * c);
  }
}

extern "C" void kernel_launch(void* const* d_in, const int* in_sizes, int n_in,
                              void* d_out, int out_size, void* d_ws, size_t ws_size,
                              hipStream_t stream) {
  (void)in_sizes; (void)n_in; (void)d_ws; (void)ws_size; (void)out_size;
  const float* x = (const float*)d_in[0];
  float* out = (float*)d_out;
  dim3 grid(NROWS / BT, NROWS / BT);  // 64 x 64 tiles
  pairwise_l1_kernel<<<grid, 256, 0, stream>>>(x, out);
}